// splineGCN_16286515986687
// MI455X (gfx1250) — compile-verified
//
#include <hip/hip_runtime.h>
#include <math.h>

typedef __attribute__((ext_vector_type(2))) float v2f;
typedef __attribute__((ext_vector_type(8))) float v8f;

// ---------------------------------------------------------------------------
// Zero-fill kernel (re-initializes atomic accumulators every call)
// ---------------------------------------------------------------------------
__global__ __launch_bounds__(256) void sg_zero(float* __restrict__ p, long long count) {
    long long t = (long long)blockIdx.x * 256 + threadIdx.x;
    if (t < count) p[t] = 0.0f;
}

// ---------------------------------------------------------------------------
// Fused triple GEMM via V_WMMA_F32_16X16X4_F32 (wave32, fp32 in / fp32 acc).
//   H0 = X @ W0 ; H1 = X @ W1 ; HR = X @ Wr + bias
// X: n x 32 (row major). W*: 32 x NF (row major). Outputs: n x NS (NS padded).
//
// Weights staged once per block into LDS, transposed + zero-padded:
//   lw[w][c*LST + kk] = (c < NF) ? W[kk*NF + c] : 0
// Column stride LST=36 floats => inner-loop ds_load_b64 is bank-conflict-free
// (lanes 0-15 hit bank pairs {4m,4m+1}, lanes 16-31 hit {4m+2,4m+3}).
// One wave owns a 16-row tile of X; K=32 -> 8 WMMA ops per 16x16 out tile.
// NF/NS are template params: column-tile loop unrolls, the pad predicate
// constant-folds, and full 16-row tiles take a store path with at most one
// exec toggle (no per-store saveexec ladder).
// ---------------------------------------------------------------------------
template <int NF, int NS>
__global__ __launch_bounds__(256) void sg_gemm3(
    const float* __restrict__ X,
    const float* __restrict__ W0, const float* __restrict__ W1,
    const float* __restrict__ Wr, const float* __restrict__ bias,
    float* __restrict__ H0, float* __restrict__ H1, float* __restrict__ HR,
    int n)
{
    constexpr int LST    = 36;                // LDS column stride (floats)
    constexpr int NTILES = (NF + 15) / 16;    // 2 (NF=32) or 1 (NF=10)
    __shared__ float lw[3 * 32 * LST];        // 3 weights, 32 cols x 32 K (13.5 KB)

    // ---- cooperative staging: transpose + zero-pad (global reads coalesced)
    {
        const float* Ws[3] = { W0, W1, Wr };
        for (int w = 0; w < 3; ++w) {
            for (int idx = threadIdx.x; idx < 32 * 32; idx += 256) {
                const int c  = idx & 31;      // output column (fast -> coalesced)
                const int kk = idx >> 5;      // K index
                lw[w * 32 * LST + c * LST + kk] =
                    (c < NF) ? Ws[w][(size_t)kk * NF + c] : 0.0f;
            }
        }
    }
    __syncthreads();

    const int lane = threadIdx.x & 31;
    const int wave = threadIdx.x >> 5;
    const int tile = blockIdx.x * 8 + wave;
    const int row0 = tile << 4;
    if (row0 >= n) return;                    // wave-uniform: EXEC stays all-ones

    const int  r        = lane & 15;          // 0..15
    const int  hi       = lane >> 4;          // 0: lanes 0-15, 1: lanes 16-31
    const int  kbase    = hi * 2;             // K sub-offset in each 4-wide step
    const int  row      = row0 + r;
    const int  rowc     = (row < n) ? row : (n - 1);  // clamp: no divergent loads
    const bool fullTile = (row0 + 16 <= n);   // wave-uniform

    // ---- A fragments: 16x4 tiles, 8 K-steps covering K=32 -----------------
    v2f a[8];
    {
        const float* xrow = X + (size_t)rowc * 32 + kbase;
#pragma unroll
        for (int k = 0; k < 8; ++k) a[k] = *(const v2f*)(xrow + k * 4);
    }

    // ---- one weight matrix -> one output ---------------------------------
    auto run = [&](int w, const float* __restrict__ bvec, float* __restrict__ H) {
        const float* lsrc = &lw[w * 32 * LST];
#pragma unroll
        for (int t = 0; t < NTILES; ++t) {
            const int c = t * 16 + r;         // c < NS always (padded cols = 0)
            v8f acc = {};
#pragma unroll
            for (int k = 0; k < 8; ++k) {
                const v2f bf = *(const v2f*)&lsrc[c * LST + k * 4 + kbase];
                acc = __builtin_amdgcn_wmma_f32_16x16x4_f32(
                        /*neg_a=*/false, a[k], /*neg_b=*/false, bf,
                        /*c_mod=*/(short)0, acc,
                        /*reuse_a=*/false, /*reuse_b=*/false);
            }
            const bool  cok = (NF % 16 == 0) ? true : (c < NF);
            const float bv  = (bvec != nullptr && cok) ? bvec[c] : 0.0f;
            float* __restrict__ Hc = H + (size_t)(row0 + hi * 8) * NS + c;

            if (fullTile) {                   // uniform branch; hot path
                if (NF % 16 == 0) {           // no pad: fully unconditional burst
#pragma unroll
                    for (int v = 0; v < 8; ++v) Hc[(size_t)v * NS] = acc[v] + bv;
                } else if (cok) {             // one exec toggle for all 8 stores
#pragma unroll
                    for (int v = 0; v < 8; ++v) Hc[(size_t)v * NS] = acc[v] + bv;
                }
            } else {                          // ragged boundary (unused for n%16==0)
#pragma unroll
                for (int v = 0; v < 8; ++v) {
                    const int rr = row0 + v + hi * 8;
                    if (cok && rr < n) H[(size_t)rr * NS + c] = acc[v] + bv;
                }
            }
        }
    };

    run(0, nullptr, H0);
    run(1, nullptr, H1);
    run(2, bias,    HR);
}

// ---------------------------------------------------------------------------
// Edge scatter: one (1<<LOGW)-lane group per edge, lane f handles feature f.
//   msg = (1-u)*h0[src] + u*h1[src];  agg[dst] += msg;  deg[dst] += 1
// ---------------------------------------------------------------------------
template <int LOGW, int NF, bool DODEG>
__global__ __launch_bounds__(256) void sg_edge_scatter(
    const int* __restrict__ ei, const float* __restrict__ ea,
    const float* __restrict__ h0, const float* __restrict__ h1,
    float* __restrict__ agg, float* __restrict__ deg, int E)
{
    constexpr int NS = 1 << LOGW;
    const long long tid = (long long)blockIdx.x * 256 + threadIdx.x;
    const int e = (int)(tid >> LOGW);
    if (e >= E) return;
    const int f = (int)tid & (NS - 1);

    const int   src = ei[e];
    const int   dst = ei[(size_t)E + e];
    const float u   = ea[e];

    if (f < NF) {
        const size_t sb = (size_t)src * NS + f;
        const float  m  = (1.0f - u) * h0[sb] + u * h1[sb];
        atomicAdd(agg + (size_t)dst * NS + f, m);
    }
    if (DODEG && f == 0) atomicAdd(deg + dst, 1.0f);
}

// ---------------------------------------------------------------------------
// Layer-1 combine: hidden = relu(agg/max(deg,1) + rootb)
// ---------------------------------------------------------------------------
__global__ __launch_bounds__(256) void sg_combine_relu(
    const float* __restrict__ agg, const float* __restrict__ deg,
    const float* __restrict__ rootb, float* __restrict__ hout, int n)
{
    const long long t = (long long)blockIdx.x * 256 + threadIdx.x;
    const int node = (int)(t >> 5);
    if (node >= n) return;
    const int f = (int)t & 31;
    const float d = fmaxf(deg[node], 1.0f);
    const size_t i = (size_t)node * 32 + f;
    const float v = agg[i] / d + rootb[i];
    hout[i] = fmaxf(v, 0.0f);
}

// ---------------------------------------------------------------------------
// Layer-2 combine + log_softmax over 10 classes (one thread per node)
// ---------------------------------------------------------------------------
__global__ __launch_bounds__(256) void sg_combine_logsoftmax(
    const float* __restrict__ agg, const float* __restrict__ deg,
    const float* __restrict__ rootb, float* __restrict__ out, int n)
{
    const int node = blockIdx.x * 256 + threadIdx.x;
    if (node >= n) return;
    const float d = fmaxf(deg[node], 1.0f);
    float v[10];
    float mx = -3.402823466e38f;
#pragma unroll
    for (int j = 0; j < 10; ++j) {
        v[j] = agg[(size_t)node * 16 + j] / d + rootb[(size_t)node * 16 + j];
        mx = fmaxf(mx, v[j]);
    }
    float s = 0.0f;
#pragma unroll
    for (int j = 0; j < 10; ++j) s += expf(v[j] - mx);
    const float lse = mx + logf(s);
#pragma unroll
    for (int j = 0; j < 10; ++j) out[(size_t)node * 10 + j] = v[j] - lse;
}

// ---------------------------------------------------------------------------
// Host-side launcher
// ---------------------------------------------------------------------------
extern "C" void kernel_launch(void* const* d_in, const int* in_sizes, int n_in,
                              void* d_out, int out_size, void* d_ws, size_t ws_size,
                              hipStream_t stream)
{
    const float* x     = (const float*)d_in[0];   // N x 32
    const int*   ei    = (const int*)  d_in[1];   // 2 x E (int32)
    const float* ea    = (const float*)d_in[2];   // E x 1
    const float* W1    = (const float*)d_in[3];   // 2 x 32 x 32
    const float* root1 = (const float*)d_in[4];   // 32 x 32
    const float* b1    = (const float*)d_in[5];   // 32
    const float* W2    = (const float*)d_in[6];   // 2 x 32 x 10
    const float* root2 = (const float*)d_in[7];   // 32 x 10
    const float* b2    = (const float*)d_in[8];   // 10
    float*       out   = (float*)d_out;

    const int N = in_sizes[0] / 32;
    const int E = in_sizes[2];
    const size_t Ns = (size_t)N;

    // Workspace layout (floats): agg | deg | h0 | h1 | rootb | hidden
    float* ws     = (float*)d_ws;
    float* agg    = ws;                 // N*32
    float* deg    = agg   + Ns * 32;    // N
    float* h0     = deg   + Ns;         // N*32
    float* h1     = h0    + Ns * 32;    // N*32
    float* rootb  = h1    + Ns * 32;    // N*32
    float* hidden = rootb + Ns * 32;    // N*32

    const int gemmBlocks = (int)((((Ns + 15) / 16) + 7) / 8);

    // ===================== Layer 1 (32 -> 32, ReLU) ========================
    {
        const long long zc = Ns * 33;                      // agg + deg contiguous
        sg_zero<<<(int)((zc + 255) / 256), 256, 0, stream>>>(agg, zc);

        sg_gemm3<32, 32><<<gemmBlocks, 256, 0, stream>>>(
            x, W1, W1 + 32 * 32, root1, b1, h0, h1, rootb, N);

        const long long et = (long long)E << 5;            // 32 lanes / edge
        sg_edge_scatter<5, 32, true><<<(int)((et + 255) / 256), 256, 0, stream>>>(
            ei, ea, h0, h1, agg, deg, E);

        const long long ct = Ns * 32;
        sg_combine_relu<<<(int)((ct + 255) / 256), 256, 0, stream>>>(
            agg, deg, rootb, hidden, N);
    }

    // ===================== Layer 2 (32 -> 10, log_softmax) =================
    {
        const long long zc = Ns * 16;
        sg_zero<<<(int)((zc + 255) / 256), 256, 0, stream>>>(agg, zc);

        sg_gemm3<10, 16><<<gemmBlocks, 256, 0, stream>>>(
            hidden, W2, W2 + 32 * 10, root2, b2, h0, h1, rootb, N);

        const long long et = (long long)E << 4;            // 16 lanes / edge
        sg_edge_scatter<4, 10, false><<<(int)((et + 255) / 256), 256, 0, stream>>>(
            ei, ea, h0, h1, agg, deg, E);

        sg_combine_logsoftmax<<<(N + 255) / 256, 256, 0, stream>>>(
            agg, deg, rootb, out, N);
    }
}